// GraphAttnLayer_21869973471559
// MI455X (gfx1250) — compile-verified
//
#include <hip/hip_runtime.h>
#include <math.h>

typedef __attribute__((ext_vector_type(2))) float v2f;
typedef __attribute__((ext_vector_type(8))) float v8f;

#define NEG_SLOPE 0.2f
#define FEAT 64
// order-preserving encoding of -inf (bits 0xFF800000, negative -> ~bits)
#define MENC_NEGINF 0x007FFFFFu

// ---- order-preserving float <-> uint mapping (unsigned compare == float compare)
__device__ __forceinline__ unsigned enc_f32(float f) {
  unsigned u = __float_as_uint(f);
  return (u & 0x80000000u) ? ~u : (u | 0x80000000u);
}
__device__ __forceinline__ float dec_f32(unsigned k) {
  unsigned u = (k & 0x80000000u) ? (k ^ 0x80000000u) : ~k;
  return __uint_as_float(u);
}

// ---------------- init per-pass accumulators ----------------
__global__ void gat_init(float* __restrict__ hacc, float* __restrict__ denom,
                         unsigned* __restrict__ menc, int N, int NF) {
  int i = blockIdx.x * blockDim.x + threadIdx.x;
  int stride = gridDim.x * blockDim.x;
  for (int idx = i; idx < NF; idx += stride) hacc[idx] = 0.0f;
  for (int idx = i; idx < N; idx += stride) {
    denom[idx] = 0.0f;
    menc[idx] = MENC_NEGINF;
  }
}

// ---------------- node scores: s[n] = dot(concat(h,q)[n], w) via f32 WMMA ----
// One wave computes 16 node scores. A-tile = 16 nodes x 4 features (f32),
// B-tile = w chunk replicated across all 16 N-columns, accumulate over K=128.
// A layout (16x4 f32): lanes 0-15 hold M=lane, K=0..1 in v0,v1; lanes 16-31
// hold K=2..3. B layout mirrors (lanes hold N, VGPR+lane-half hold K); since
// B is column-replicated only the K mapping matters.
__global__ void gat_scores_wmma(const float* __restrict__ h,
                                const float* __restrict__ q,
                                const float* __restrict__ w,
                                float* __restrict__ s, int N) {
  const int lane = threadIdx.x & 31;
  const int wave = threadIdx.x >> 5;
  const int tile = blockIdx.x * (blockDim.x >> 5) + wave;  // 16 nodes per tile
  const int node_base = tile * 16;
  if (node_base >= N) return;  // uniform across the wave

  const int mrow = lane & 15;
  const int khalf = (lane >> 4) << 1;  // 0 or 2
  const int node = node_base + mrow;
  const bool valid = node < N;
  const int nsafe = valid ? node : (N - 1);
  const float* __restrict__ hrow = h + (size_t)nsafe * FEAT;
  const float* __restrict__ qrow = q + (size_t)nsafe * FEAT;

  v8f acc = {0.f, 0.f, 0.f, 0.f, 0.f, 0.f, 0.f, 0.f};
#pragma unroll
  for (int k0 = 0; k0 < 2 * FEAT; k0 += 4) {
    const int k = k0 + khalf;  // pairs never straddle the h/q boundary (64 % 4 == 0)
    float ax = (k < FEAT) ? hrow[k] : qrow[k - FEAT];
    float ay = (k < FEAT) ? hrow[k + 1] : qrow[k + 1 - FEAT];
    if (!valid) { ax = 0.f; ay = 0.f; }
    v2f A = {ax, ay};
    v2f B = {w[k], w[k + 1]};  // replicate w across all 16 output columns
    acc = __builtin_amdgcn_wmma_f32_16x16x4_f32(
        /*neg_a=*/false, A, /*neg_b=*/false, B,
        /*c_mod=*/(short)0, acc, /*reuse_a=*/false, /*reuse_b=*/false);
  }
  // C layout: VGPR r, lanes 0-15 -> (M=r, N=lane); lanes 16-31 -> (M=8+r).
  // Column N=0 lives in lanes 0 and 16.
  if (mrow == 0) {
    const int base = node_base + ((lane >> 4) << 3);
#pragma unroll
    for (int r = 0; r < 8; ++r) {
      const int n2 = base + r;
      if (n2 < N) s[n2] = acc[r];
    }
  }
}

// ---------------- edge pass 1: leaky-relu score + segment max ---------------
__global__ void gat_edge_max(const float* __restrict__ s,
                             const int* __restrict__ src,
                             const int* __restrict__ dst,
                             float* __restrict__ ew,
                             unsigned* __restrict__ menc, int E) {
  const int e = blockIdx.x * blockDim.x + threadIdx.x;
  if (e >= E) return;
  const float sv = s[src[e]];
  const float a = sv > 0.f ? sv : NEG_SLOPE * sv;
  ew[e] = a;
  atomicMax(&menc[dst[e]], enc_f32(a));
}

// ---------------- edge pass 2: exp(a - m[dst]) + segment sum ---------------
__global__ void gat_edge_exp(float* __restrict__ ew,
                             const int* __restrict__ dst,
                             const unsigned* __restrict__ menc,
                             float* __restrict__ denom, int E) {
  const int e = blockIdx.x * blockDim.x + threadIdx.x;
  if (e >= E) return;
  const int d = dst[e];
  const float ex = expf(ew[e] - dec_f32(menc[d]));
  ew[e] = ex;  // overwrite a with ex
  atomicAdd(&denom[d], ex);
}

// ---------------- edge pass 3: hacc[dst] += (ex/denom[dst]) * h[src] -------
// One wave per edge; lane covers 2 of the 64 features (float2, coalesced).
__global__ void gat_edge_agg(const float* __restrict__ hin,
                             const float* __restrict__ ew,
                             const float* __restrict__ denom,
                             const int* __restrict__ src,
                             const int* __restrict__ dst,
                             float* __restrict__ hacc, int E) {
  const int t = blockIdx.x * blockDim.x + threadIdx.x;
  const int e = t >> 5;
  const int lane = t & 31;
  if (e >= E) return;
  const int d = dst[e];
  const float alpha = ew[e] / denom[d];
  const float2* __restrict__ hr =
      (const float2*)(hin + (size_t)src[e] * FEAT);
  const float2 v = hr[lane];
  float* o = hacc + (size_t)d * FEAT + lane * 2;
  atomicAdd(o, alpha * v.x);
  atomicAdd(o + 1, alpha * v.y);
}

// ---------------- finalize: keep old row where node has no in-edges --------
__global__ void gat_finalize(const float* __restrict__ hacc,
                             const float* __restrict__ denom,
                             const float* __restrict__ hin,
                             float* __restrict__ hout, int NF) {
  const int i = blockIdx.x * blockDim.x + threadIdx.x;
  if (i >= NF) return;
  const int n = i >> 6;  // FEAT == 64
  hout[i] = (denom[n] > 0.f) ? hacc[i] : hin[i];
}

extern "C" void kernel_launch(void* const* d_in, const int* in_sizes, int n_in,
                              void* d_out, int out_size, void* d_ws,
                              size_t ws_size, hipStream_t stream) {
  const float* h = (const float*)d_in[0];
  const float* q = (const float*)d_in[1];
  const int* src = (const int*)d_in[2];
  const int* dst = (const int*)d_in[3];
  const float* w1 = (const float*)d_in[4];
  const float* w2 = (const float*)d_in[5];

  const int N = in_sizes[0] / FEAT;
  const int E = in_sizes[2];
  const int NF = N * FEAT;

  // workspace carve-up (all float-sized slots)
  float* s = (float*)d_ws;                 // N
  unsigned* menc = (unsigned*)(s + N);     // N
  float* denom = (float*)(menc + N);       // N
  float* ew = denom + N;                   // E
  float* h1 = ew + E;                      // N*FEAT (pass-1 output)
  float* hacc = h1 + (size_t)NF;           // N*FEAT (accumulator)

  const int TB = 256;
  const int initBlocks = 1024;
  const int waves = (N + 15) / 16;
  const int scoreBlocks = (waves + 7) / 8;  // 8 waves (of 32) per 256-thr block
  const int edgeBlocks = (E + TB - 1) / TB;
  const int aggBlocks = (int)(((long long)E * 32 + TB - 1) / TB);
  const int nfBlocks = (NF + TB - 1) / TB;

  // ---------------- pass 1: h -> h1 with w1 ----------------
  gat_init<<<initBlocks, TB, 0, stream>>>(hacc, denom, menc, N, NF);
  gat_scores_wmma<<<scoreBlocks, TB, 0, stream>>>(h, q, w1, s, N);
  gat_edge_max<<<edgeBlocks, TB, 0, stream>>>(s, src, dst, ew, menc, E);
  gat_edge_exp<<<edgeBlocks, TB, 0, stream>>>(ew, dst, menc, denom, E);
  gat_edge_agg<<<aggBlocks, TB, 0, stream>>>(h, ew, denom, src, dst, hacc, E);
  gat_finalize<<<nfBlocks, TB, 0, stream>>>(hacc, denom, h, h1, NF);

  // ---------------- pass 2: h1 -> d_out with w2 ----------------
  float* out = (float*)d_out;
  gat_init<<<initBlocks, TB, 0, stream>>>(hacc, denom, menc, N, NF);
  gat_scores_wmma<<<scoreBlocks, TB, 0, stream>>>(h1, q, w2, s, N);
  gat_edge_max<<<edgeBlocks, TB, 0, stream>>>(s, src, dst, ew, menc, E);
  gat_edge_exp<<<edgeBlocks, TB, 0, stream>>>(ew, dst, menc, denom, E);
  gat_edge_agg<<<aggBlocks, TB, 0, stream>>>(h1, ew, denom, src, dst, hacc, E);
  gat_finalize<<<nfBlocks, TB, 0, stream>>>(hacc, denom, h1, out, NF);
}